// RNN_60129542175
// MI455X (gfx1250) — compile-verified
//
#include <hip/hip_runtime.h>
#include <cmath>

typedef __attribute__((ext_vector_type(2))) float v2f;
typedef __attribute__((ext_vector_type(8))) float v8f;

constexpr int B_ = 64, T_ = 512, I_ = 1024, H_ = 1024;
constexpr int M_ = B_ * T_;              // flattened (b,t) rows of the big GEMM

// D = A(16x4 f32) x B(4x16 f32) + C(16x16 f32), full-precision matrix path.
__device__ __forceinline__ v8f wmma4(v2f a, v2f b, v8f c) {
  return __builtin_amdgcn_wmma_f32_16x16x4_f32(
      /*neg_a=*/false, a, /*neg_b=*/false, b,
      /*c_mod=*/(short)0, c, /*reuse_a=*/false, /*reuse_b=*/false);
}

// ---------------------------------------------------------------------------
// Kernel 1: x_proj[m, n] = sum_k seq[m, k] * W_ih[n, k] + b_ih[n] + b_hh[n]
//   m in [0, B*T), n in [0, H), k in [0, I)
// One wave owns a 16(M) x 64(N) tile: 4 f32-WMMA accumulators, A-frag reused 4x.
// 4 waves per block share the same N-group -> B loads hit L0/L2.
// ---------------------------------------------------------------------------
__global__ __launch_bounds__(128) void xproj_wmma(
    const float* __restrict__ seq, const float* __restrict__ W_ih,
    const float* __restrict__ b_ih, const float* __restrict__ b_hh,
    float* __restrict__ xp)
{
  const int lane = threadIdx.x & 31;
  const int wave = threadIdx.x >> 5;
  const int half = lane >> 4;        // lanes 16-31 hold K+2
  const int lm   = lane & 15;
  const int n0 = blockIdx.x * 64;
  const int m0 = (blockIdx.y * 4 + wave) * 16;

  const float* a_row  = seq  + (size_t)(m0 + lm) * I_ + 2 * half;
  const float* b_row0 = W_ih + (size_t)(n0 +  0 + lm) * I_ + 2 * half;
  const float* b_row1 = W_ih + (size_t)(n0 + 16 + lm) * I_ + 2 * half;
  const float* b_row2 = W_ih + (size_t)(n0 + 32 + lm) * I_ + 2 * half;
  const float* b_row3 = W_ih + (size_t)(n0 + 48 + lm) * I_ + 2 * half;

  v8f c0 = {}, c1 = {}, c2 = {}, c3 = {};
  for (int k = 0; k < I_; k += 4) {
    v2f a  = *(const v2f*)(a_row  + k);
    v2f b0 = *(const v2f*)(b_row0 + k);
    v2f b1 = *(const v2f*)(b_row1 + k);
    v2f b2 = *(const v2f*)(b_row2 + k);
    v2f b3 = *(const v2f*)(b_row3 + k);
    c0 = wmma4(a, b0, c0);
    c1 = wmma4(a, b1, c1);
    c2 = wmma4(a, b2, c2);
    c3 = wmma4(a, b3, c3);
  }

  const float bias0 = b_ih[n0 +  0 + lm] + b_hh[n0 +  0 + lm];
  const float bias1 = b_ih[n0 + 16 + lm] + b_hh[n0 + 16 + lm];
  const float bias2 = b_ih[n0 + 32 + lm] + b_hh[n0 + 32 + lm];
  const float bias3 = b_ih[n0 + 48 + lm] + b_hh[n0 + 48 + lm];

  #pragma unroll
  for (int v = 0; v < 8; ++v) {
    const int row = m0 + v + 8 * half;          // C layout: VGPR v -> rows v, v+8
    float* o = xp + (size_t)row * H_ + n0;
    o[ 0 + lm] = c0[v] + bias0;
    o[16 + lm] = c1[v] + bias1;
    o[32 + lm] = c2[v] + bias2;
    o[48 + lm] = c3[v] + bias3;
  }
}

// ---------------------------------------------------------------------------
// Kernel 2 (one launch per time step t):
//   h_t[b, n] = tanh(xp[b, t, n] + sum_k h_{t-1}[b, k] * W_hh[n, k])
// h_{t-1} is read directly from all_h[:, t-1, :] (t==0 -> zeros).
// Block = 16 waves = 4 M-tiles x split-K=4 (chains of 256 WMMAs/wave),
// partials reduced through LDS, then tanh + store. 16 blocks cover N=1024.
// ---------------------------------------------------------------------------
__global__ __launch_bounds__(512) void rnn_step_wmma(
    const float* __restrict__ xp, const float* __restrict__ W_hh,
    float* __restrict__ all_h, float* __restrict__ last_h, int t)
{
  __shared__ float red[4][4][16 * 64];   // [m-tile][k-chunk][16x64 partial] = 64 KB

  const int lane = threadIdx.x & 31;
  const int wave = threadIdx.x >> 5;
  const int half = lane >> 4;
  const int lm   = lane & 15;
  const int mt = wave & 3;               // which 16 batch rows
  const int kc = wave >> 2;              // which K chunk of 256
  const int n0 = blockIdx.x * 64;
  const int m0 = mt * 16;
  const int kbeg = kc * (H_ / 4);

  v8f c0 = {}, c1 = {}, c2 = {}, c3 = {};
  if (t > 0) {
    const float* a_row  = all_h + ((size_t)(m0 + lm) * T_ + (t - 1)) * H_ + 2 * half;
    const float* b_row0 = W_hh + (size_t)(n0 +  0 + lm) * H_ + 2 * half;
    const float* b_row1 = W_hh + (size_t)(n0 + 16 + lm) * H_ + 2 * half;
    const float* b_row2 = W_hh + (size_t)(n0 + 32 + lm) * H_ + 2 * half;
    const float* b_row3 = W_hh + (size_t)(n0 + 48 + lm) * H_ + 2 * half;
    for (int k = kbeg; k < kbeg + H_ / 4; k += 4) {
      v2f a  = *(const v2f*)(a_row  + k);
      v2f b0 = *(const v2f*)(b_row0 + k);
      v2f b1 = *(const v2f*)(b_row1 + k);
      v2f b2 = *(const v2f*)(b_row2 + k);
      v2f b3 = *(const v2f*)(b_row3 + k);
      c0 = wmma4(a, b0, c0);
      c1 = wmma4(a, b1, c1);
      c2 = wmma4(a, b2, c2);
      c3 = wmma4(a, b3, c3);
    }
  }

  // Stash this wave's 16x64 partial into LDS.
  float* my = &red[mt][kc][0];
  #pragma unroll
  for (int v = 0; v < 8; ++v) {
    const int r = v + 8 * half;
    my[r * 64 +  0 + lm] = c0[v];
    my[r * 64 + 16 + lm] = c1[v];
    my[r * 64 + 32 + lm] = c2[v];
    my[r * 64 + 48 + lm] = c3[v];
  }
  __syncthreads();

  // One wave per M-tile reduces the 4 K-chunks, applies bias'd xp + tanh, stores.
  if (kc == 0) {
    for (int idx = lane; idx < 16 * 64; idx += 32) {
      const float s = red[mt][0][idx] + red[mt][1][idx] +
                      red[mt][2][idx] + red[mt][3][idx];
      const int r  = idx >> 6;
      const int cg = n0 + (idx & 63);
      const int m  = m0 + r;                              // batch index
      const size_t off = ((size_t)m * T_ + t) * H_ + cg;  // [B,T,H]
      const float h = tanhf(xp[off] + s);
      all_h[off] = h;
      if (t == T_ - 1) last_h[(size_t)m * H_ + cg] = h;
    }
  }
}

extern "C" void kernel_launch(void* const* d_in, const int* in_sizes, int n_in,
                              void* d_out, int out_size, void* d_ws, size_t ws_size,
                              hipStream_t stream) {
  (void)in_sizes; (void)n_in; (void)out_size; (void)ws_size;
  const float* seq  = (const float*)d_in[0];
  const float* W_ih = (const float*)d_in[1];
  const float* W_hh = (const float*)d_in[2];
  const float* b_ih = (const float*)d_in[3];
  const float* b_hh = (const float*)d_in[4];

  float* all_h  = (float*)d_out;                      // [B,T,H]
  float* last_h = all_h + (size_t)B_ * T_ * H_;       // [B,H]
  float* xp     = (float*)d_ws;                       // [B,T,H] = 128 MB scratch

  // Phase 1: one big fp32 WMMA GEMM (biases folded in).
  dim3 g1(H_ / 64, M_ / (4 * 16));                    // (16, 512), 128 thr/block
  xproj_wmma<<<g1, 128, 0, stream>>>(seq, W_ih, b_ih, b_hh, xp);

  // Phase 2: 512 dependent steps; stream order provides the global sync.
  for (int t = 0; t < T_; ++t)
    rnn_step_wmma<<<dim3(H_ / 64), 512, 0, stream>>>(xp, W_hh, all_h, last_h, t);
}